// FoldIteration_88622355185763
// MI455X (gfx1250) — compile-verified
//
#include <hip/hip_runtime.h>
#include <math.h>

typedef __attribute__((ext_vector_type(2))) float v2f;
typedef __attribute__((ext_vector_type(8))) float v8f;

#define NRES 768
#define C1   384
#define CC2  128
#define NH   12
#define PVP  8
#define NN   (NRES*NRES)
#define FCOL 2112   // 192 + 4*96 + 1536
#define OCOL 396    // 384 + 9 + 3

__device__ __forceinline__ v8f wmma4(v2f a, v2f b, v8f c) {
  // V_WMMA_F32_16X16X4_F32 : D = A(16x4) * B(4x16) + C(16x16)
  return __builtin_amdgcn_wmma_f32_16x16x4_f32(false, a, false, b, (short)0, c,
                                               false, false);
}

// ---------------------------------------------------------------------------
// Tensor Data Mover: 1-D f32 tile load Global -> LDS (D# per ISA 08 §8.3/8.4)
// ---------------------------------------------------------------------------
#if defined(__AMDGCN__) && __has_builtin(__builtin_amdgcn_tensor_load_to_lds) && \
    __has_builtin(__builtin_amdgcn_s_wait_tensorcnt)
#define USE_TDM 1
#else
#define USE_TDM 0
#endif

#if USE_TDM
typedef __attribute__((ext_vector_type(4))) unsigned int u32x4;
typedef __attribute__((ext_vector_type(8))) int i32x8;
typedef __attribute__((ext_vector_type(4))) int i32x4;

__device__ __forceinline__ unsigned lds_off(const void* p) {
  // LDS aperture: flat addr low 32 bits are the LDS byte offset
  return (unsigned)(size_t)p;
}

// Contiguous 1-D tile of `nelem` f32 (nelem < 65536), Global -> LDS.
__device__ __forceinline__ void tdm_load_1d(unsigned lds_byte,
                                            const float* gptr,
                                            unsigned nelem) {
  const unsigned long long ga = (unsigned long long)(size_t)gptr;
  u32x4 g0;
  g0[0] = 1u;                                    // count=1 (valid), user mode
  g0[1] = lds_byte;                              // lds_addr
  g0[2] = (unsigned)(ga & 0xffffffffu);          // global_addr[31:0]
  g0[3] = (unsigned)((ga >> 32) & 0x01ffffffu)   // global_addr[56:32]
        | (2u << 30);                            // type = 2 ("image")
  i32x8 g1;
  g1[0] = (int)(2u << 16);                       // data_size = 4B, mask=0
  g1[1] = (int)((nelem & 0xffffu) << 16);        // tensor_dim0[15:0]
  g1[2] = (int)(((nelem >> 16) & 0xffffu)        // tensor_dim0[31:16]
        | (1u << 16));                           // tensor_dim1 = 1
  g1[3] = (int)((nelem & 0xffffu) << 16);        // tile_dim0 = nelem
  g1[4] = 0;                                     // tile_dim1/2 unused
  g1[5] = (int)nelem;                            // tensor_dim0_stride[31:0]
  g1[6] = 0;
  g1[7] = 0;
  i32x4 z4 = {0, 0, 0, 0};
#if __clang_major__ >= 23
  i32x8 z8 = {0, 0, 0, 0, 0, 0, 0, 0};
  __builtin_amdgcn_tensor_load_to_lds(g0, g1, z4, z4, z8, 0);
#else
  __builtin_amdgcn_tensor_load_to_lds(g0, g1, z4, z4, 0);
#endif
}
#endif  // USE_TDM

// ---------------------------------------------------------------------------
// Generic row-major WMMA-f32 GEMM: C[M,Nc] = A[M,K] @ B[K,Nc] (+bias)(+relu)
// M, Nc multiples of 16; K multiple of 4. One 16x16 tile per wave.
// ---------------------------------------------------------------------------
__global__ void k_gemm(const float* __restrict__ A, const float* __restrict__ B,
                       float* __restrict__ C, int M, int Nc, int K,
                       const float* __restrict__ bias, int relu) {
  const int lane = threadIdx.x & 31;
  const int wave = threadIdx.x >> 5;
  const int nT = Nc >> 4;
  const int tile = blockIdx.x * 8 + wave;
  if (tile >= (M >> 4) * nT) return;                 // wave-uniform
  const int mt = tile / nT, nt = tile % nT;
  const int r = lane & 15, half = lane >> 4;
  const float* ap = A + (size_t)(mt * 16 + r) * K + 2 * half;
  const float* bp = B + (size_t)(2 * half) * Nc + (nt * 16 + r);
  const size_t bstep = (size_t)4 * Nc;
  v8f acc = {0.f, 0.f, 0.f, 0.f, 0.f, 0.f, 0.f, 0.f};
  for (int k0 = 0; k0 < K; k0 += 4) {
    v2f a, b;
    a.x = ap[0];
    a.y = ap[1];
    b.x = bp[0];
    b.y = bp[Nc];
    acc = wmma4(a, b, acc);
    ap += 4;
    bp += bstep;
  }
  const int col = nt * 16 + r;
  const float bv = bias ? bias[col] : 0.f;
  float* Cp = C + (size_t)(mt * 16 + 8 * half) * Nc + col;
#pragma unroll
  for (int j = 0; j < 8; ++j) {
    float v = acc[j] + bv;
    if (relu) v = fmaxf(v, 0.f);
    Cp[(size_t)j * Nc] = v;
  }
}

// ---------------------------------------------------------------------------
// Rigid apply: raw[n, h*3P..] (x|y|z thirds) -> global pts g[n,(h*P+p)*3+d],
// optional squared-norm accumulation norms[n*NH+h].
// ---------------------------------------------------------------------------
__global__ void k_rigid_fwd(const float* __restrict__ raw,
                            const float* __restrict__ rot,
                            const float* __restrict__ trans,
                            float* __restrict__ g, float* __restrict__ norms,
                            int P) {
  const int idx = blockIdx.x * blockDim.x + threadIdx.x;  // n*NH + h
  if (idx >= NRES * NH) return;
  const int n = idx / NH, h = idx % NH;
  const float* R = rot + n * 9;
  const float* t = trans + n * 3;
  const float* pr = raw + (size_t)n * (NH * 3 * P) + h * 3 * P;
  float* go = g + (size_t)n * (NH * P * 3) + h * P * 3;
  float s = 0.f;
  for (int p = 0; p < P; ++p) {
    const float lx = pr[p], ly = pr[P + p], lz = pr[2 * P + p];
    const float gx = R[0] * lx + R[1] * ly + R[2] * lz + t[0];
    const float gy = R[3] * lx + R[4] * ly + R[5] * lz + t[1];
    const float gz = R[6] * lx + R[7] * ly + R[8] * lz + t[2];
    go[p * 3 + 0] = gx;
    go[p * 3 + 1] = gy;
    go[p * 3 + 2] = gz;
    s += gx * gx + gy * gy + gz * gz;
  }
  if (norms) norms[idx] = s;
}

// ---------------------------------------------------------------------------
// logits base: [h][q*N+k] = inputs_2d[q,k,:] @ w_2d[:,h] + b_2d[h]
// Block = 128 consecutive flat rows staged TDM->LDS (64 KB), 8 WMMA tiles.
// ---------------------------------------------------------------------------
__global__ void k_logits2d(const float* __restrict__ in2d,
                           const float* __restrict__ w2d,
                           const float* __restrict__ b2d,
                           float* __restrict__ logits) {
  __shared__ float sA[128 * CC2];  // 64 KB
  const int lane = threadIdx.x & 31;
  const int wave = threadIdx.x >> 5;
  const size_t base = (size_t)blockIdx.x * 128;  // flat (q*N+k) row base
#if USE_TDM
  if (wave == 0) {
    tdm_load_1d(lds_off(sA), in2d + base * CC2, 128 * CC2);
    __builtin_amdgcn_s_wait_tensorcnt(0);
  }
  __syncthreads();
#else
  for (int i = threadIdx.x; i < 128 * CC2; i += 256)
    sA[i] = in2d[base * CC2 + i];
  __syncthreads();
#endif
  const int r = lane & 15, half = lane >> 4;
  const float* Arow = sA + (wave * 16 + r) * CC2 + 2 * half;
  v8f acc = {0.f, 0.f, 0.f, 0.f, 0.f, 0.f, 0.f, 0.f};
  for (int k0 = 0; k0 < CC2; k0 += 4) {
    const int ka = k0 + 2 * half;
    v2f a, b;
    a.x = Arow[k0];
    a.y = Arow[k0 + 1];
    b.x = (r < NH) ? w2d[ka * NH + r] : 0.f;
    b.y = (r < NH) ? w2d[(ka + 1) * NH + r] : 0.f;
    acc = wmma4(a, b, acc);
  }
  if (r < NH) {
    const float bv = b2d[r];
    const size_t rowbase = base + (size_t)wave * 16 + 8 * half;
#pragma unroll
    for (int j = 0; j < 8; ++j)
      logits[(size_t)r * NN + rowbase + j] = acc[j] + bv;
  }
}

// ---------------------------------------------------------------------------
// Per-head K=28 GEMM: qs.ks (scaled 1/sqrt(16)) + pw*qg.kg, fused with
// -0.5*pw*(|q|^2+|k|^2), 2d base, mask and sqrt(1/3) scale.
// ---------------------------------------------------------------------------
__global__ void k_qk_terms(const float* __restrict__ qs,
                           const float* __restrict__ ks,
                           const float* __restrict__ qg,
                           const float* __restrict__ kg,
                           const float* __restrict__ qn,
                           const float* __restrict__ kn,
                           const float* __restrict__ tpw,
                           const float* __restrict__ mask,
                           float* __restrict__ logits) {
  const int lane = threadIdx.x & 31;
  const int wave = threadIdx.x >> 5;
  const int h = blockIdx.z;
  const int qt = blockIdx.y;
  const int kt = blockIdx.x * 8 + wave;
  const int r = lane & 15, half = lane >> 4;
  const float pwh = 0.23570226039f * log1pf(expf(tpw[h]));  // sqrt(2/36)*softplus
  const int qrow = qt * 16 + r;
  const int krow = kt * 16 + r;
  v8f acc = {0.f, 0.f, 0.f, 0.f, 0.f, 0.f, 0.f, 0.f};
  for (int k0 = 0; k0 < 28; k0 += 4) {
    const int c0 = k0 + 2 * half;
    v2f a, b;
    a.x = (c0 < 16) ? 0.25f * qs[(size_t)qrow * 192 + h * 16 + c0]
                    : pwh * qg[(size_t)qrow * 144 + h * 12 + (c0 - 16)];
    a.y = (c0 + 1 < 16) ? 0.25f * qs[(size_t)qrow * 192 + h * 16 + c0 + 1]
                        : pwh * qg[(size_t)qrow * 144 + h * 12 + (c0 - 15)];
    b.x = (c0 < 16) ? ks[(size_t)krow * 192 + h * 16 + c0]
                    : kg[(size_t)krow * 144 + h * 12 + (c0 - 16)];
    b.y = (c0 + 1 < 16) ? ks[(size_t)krow * 192 + h * 16 + c0 + 1]
                        : kg[(size_t)krow * 144 + h * 12 + (c0 - 15)];
    acc = wmma4(a, b, acc);
  }
  const int k = kt * 16 + r;
  const float mk = mask[k];
#pragma unroll
  for (int j = 0; j < 8; ++j) {
    const int q = qt * 16 + j + 8 * half;
    const size_t off = (size_t)h * NN + (size_t)q * NRES + k;
    float val = logits[off] + acc[j]
              - 0.5f * pwh * (qn[q * NH + h] + kn[k * NH + h])
              - 1e5f * (1.f - mask[q] * mk);
    logits[off] = val * 0.57735026919f;  // sqrt(1/3)
  }
}

// ---------------------------------------------------------------------------
// Softmax over keys for each (h,q) row, in place.
// ---------------------------------------------------------------------------
__global__ void k_softmax(float* __restrict__ logits) {
  const int q = blockIdx.x, h = blockIdx.y;
  float* row = logits + (size_t)h * NN + (size_t)q * NRES;
  const int t = threadIdx.x;  // 256 threads, 3 elems each
  __shared__ float red[8];
  const float x0 = row[t], x1 = row[t + 256], x2 = row[t + 512];
  float m = fmaxf(x0, fmaxf(x1, x2));
  for (int o = 16; o; o >>= 1) m = fmaxf(m, __shfl_xor(m, o, 32));
  if ((t & 31) == 0) red[t >> 5] = m;
  __syncthreads();
  m = red[0];
#pragma unroll
  for (int i = 1; i < 8; ++i) m = fmaxf(m, red[i]);
  __syncthreads();
  const float e0 = __expf(x0 - m), e1 = __expf(x1 - m), e2 = __expf(x2 - m);
  float s = e0 + e1 + e2;
  for (int o = 16; o; o >>= 1) s += __shfl_xor(s, o, 32);
  if ((t & 31) == 0) red[t >> 5] = s;
  __syncthreads();
  s = red[0];
#pragma unroll
  for (int i = 1; i < 8; ++i) s += red[i];
  const float inv = 1.f / s;
  row[t] = e0 * inv;
  row[t + 256] = e1 * inv;
  row[t + 512] = e2 * inv;
}

// ---------------------------------------------------------------------------
// attn @ [vs | vg] per head: K=768, Ncols = 16 + 24 = 40 (3 tiles, guarded)
// res_scalar -> final[:,0:192], rp_g -> scratch
// ---------------------------------------------------------------------------
__device__ __forceinline__ float velem(const float* vs, const float* vg, int kk,
                                       int col, int h) {
  if (col < 16) return vs[(size_t)kk * 192 + h * 16 + col];
  if (col < 40) return vg[(size_t)kk * 288 + h * 24 + (col - 16)];
  return 0.f;
}

__global__ void k_attn_v(const float* __restrict__ attn,
                         const float* __restrict__ vs,
                         const float* __restrict__ vg,
                         float* __restrict__ finalb, float* __restrict__ rp_g) {
  const int lane = threadIdx.x & 31;
  const int wave = threadIdx.x >> 5;
  const int h = blockIdx.z;
  const int tile = blockIdx.x * 8 + wave;  // 48 mtiles * 3 ntiles
  if (tile >= 144) return;
  const int mt = tile / 3, nt = tile % 3;
  const int r = lane & 15, half = lane >> 4;
  const float* Arow = attn + (size_t)h * NN + (size_t)(mt * 16 + r) * NRES;
  const int col = nt * 16 + r;
  v8f acc = {0.f, 0.f, 0.f, 0.f, 0.f, 0.f, 0.f, 0.f};
  for (int k0 = 0; k0 < NRES; k0 += 4) {
    const int ka = k0 + 2 * half;
    v2f a, b;
    a.x = Arow[ka];
    a.y = Arow[ka + 1];
    b.x = velem(vs, vg, ka, col, h);
    b.y = velem(vs, vg, ka + 1, col, h);
    acc = wmma4(a, b, acc);
  }
#pragma unroll
  for (int j = 0; j < 8; ++j) {
    const int q = mt * 16 + j + 8 * half;
    if (col < 16)
      finalb[(size_t)q * FCOL + h * 16 + col] = acc[j];
    else if (col < 40)
      rp_g[(size_t)q * 288 + h * 24 + (col - 16)] = acc[j];
  }
}

// ---------------------------------------------------------------------------
// res_2d[q,h,c] = sum_k attn[h,q,k] * inputs_2d[q,k,c]
// One block per q; double-buffered TDM stream of the [768,128] slab through
// LDS (12 chunks x 32 KB); 8 waves cover the 128 channels. M=12 pad 16.
// ---------------------------------------------------------------------------
#define RCHUNK 64
__global__ void k_res2d(const float* __restrict__ attn,
                        const float* __restrict__ in2d,
                        float* __restrict__ finalb) {
  __shared__ float sB[2][RCHUNK * CC2];  // 2 x 32 KB
  const int lane = threadIdx.x & 31;
  const int wave = threadIdx.x >> 5;  // c-tile 0..7
  const int q = blockIdx.x;
  const int r = lane & 15, half = lane >> 4;
  const int col = wave * 16 + r;
  const float* Bbase = in2d + (size_t)q * NRES * CC2;
  const float* Aro = attn + (size_t)q * NRES;  // + r*NN
  const int nch = NRES / RCHUNK;
  v8f acc = {0.f, 0.f, 0.f, 0.f, 0.f, 0.f, 0.f, 0.f};
#if USE_TDM
  if (wave == 0) {
    tdm_load_1d(lds_off(sB[0]), Bbase, RCHUNK * CC2);
    __builtin_amdgcn_s_wait_tensorcnt(0);
  }
  __syncthreads();
#endif
  for (int c = 0; c < nch; ++c) {
    const int buf = c & 1;
#if USE_TDM
    if (wave == 0 && c + 1 < nch)
      tdm_load_1d(lds_off(sB[buf ^ 1]),
                  Bbase + (size_t)(c + 1) * RCHUNK * CC2, RCHUNK * CC2);
#else
    for (int i = threadIdx.x; i < RCHUNK * CC2; i += 256)
      sB[buf][i] = Bbase[(size_t)c * RCHUNK * CC2 + i];
    __syncthreads();
#endif
    const float* Bl = sB[buf];
    for (int k0 = 0; k0 < RCHUNK; k0 += 4) {
      const int ka = k0 + 2 * half;
      const int kgl = c * RCHUNK + ka;
      v2f a, b;
      a.x = (r < NH) ? Aro[(size_t)r * NN + kgl] : 0.f;
      a.y = (r < NH) ? Aro[(size_t)r * NN + kgl + 1] : 0.f;
      b.x = Bl[ka * CC2 + col];
      b.y = Bl[(ka + 1) * CC2 + col];
      acc = wmma4(a, b, acc);
    }
#if USE_TDM
    if (wave == 0) __builtin_amdgcn_s_wait_tensorcnt(0);
#endif
    __syncthreads();
  }
#pragma unroll
  for (int j = 0; j < 8; ++j) {
    const int h = j + 8 * half;
    if (h < NH) finalb[(size_t)q * FCOL + 576 + h * CC2 + col] = acc[j];
  }
}

// ---------------------------------------------------------------------------
// rp_l = rot^T (rp_g - trans); norms; write into final[:,192:576]
// ---------------------------------------------------------------------------
__global__ void k_rp_local(const float* __restrict__ rp_g,
                           const float* __restrict__ rot,
                           const float* __restrict__ trans,
                           float* __restrict__ finalb) {
  const int idx = blockIdx.x * blockDim.x + threadIdx.x;  // n*96 + (h*8+p)
  if (idx >= NRES * NH * PVP) return;
  const int n = idx / (NH * PVP), rem = idx % (NH * PVP);
  const float* R = rot + n * 9;
  const float* t = trans + n * 3;
  const float* g = rp_g + (size_t)n * 288 + rem * 3;
  const float dx = g[0] - t[0], dy = g[1] - t[1], dz = g[2] - t[2];
  const float lx = R[0] * dx + R[3] * dy + R[6] * dz;
  const float ly = R[1] * dx + R[4] * dy + R[7] * dz;
  const float lz = R[2] * dx + R[5] * dy + R[8] * dz;
  const float nr = sqrtf(fmaxf(lx * lx + ly * ly + lz * lz, 1e-16f));
  float* f = finalb + (size_t)n * FCOL;
  f[192 + rem] = lx;
  f[288 + rem] = ly;
  f[384 + rem] = lz;
  f[480 + rem] = nr;
}

// ---------------------------------------------------------------------------
// out = LayerNorm(x (+ y)) * s + b ; optional second copy with OCOL stride
// ---------------------------------------------------------------------------
__global__ void k_add_ln(const float* __restrict__ x, const float* __restrict__ y,
                         const float* __restrict__ s, const float* __restrict__ b,
                         float* __restrict__ out, float* __restrict__ out2) {
  const int n = blockIdx.x, t = threadIdx.x;  // 128 threads, 3 elems each
  __shared__ float red[4];
  float v[3];
  float acc = 0.f;
#pragma unroll
  for (int i = 0; i < 3; ++i) {
    const int c = t + 128 * i;
    float xv = x[(size_t)n * C1 + c];
    if (y) xv += y[(size_t)n * C1 + c];
    v[i] = xv;
    acc += xv;
  }
  for (int o = 16; o; o >>= 1) acc += __shfl_xor(acc, o, 32);
  if ((t & 31) == 0) red[t >> 5] = acc;
  __syncthreads();
  const float mu = (red[0] + red[1] + red[2] + red[3]) * (1.f / C1);
  __syncthreads();
  float va = 0.f;
#pragma unroll
  for (int i = 0; i < 3; ++i) {
    const float d = v[i] - mu;
    va += d * d;
  }
  for (int o = 16; o; o >>= 1) va += __shfl_xor(va, o, 32);
  if ((t & 31) == 0) red[t >> 5] = va;
  __syncthreads();
  const float var = (red[0] + red[1] + red[2] + red[3]) * (1.f / C1);
  const float rstd = rsqrtf(var + 1e-5f);
#pragma unroll
  for (int i = 0; i < 3; ++i) {
    const int c = t + 128 * i;
    const float o_ = (v[i] - mu) * rstd * s[c] + b[c];
    out[(size_t)n * C1 + c] = o_;
    if (out2) out2[(size_t)n * OCOL + c] = o_;
  }
}

// ---------------------------------------------------------------------------
// v6 = act @ w_quat + b_quat ; quat=(1,a,b,c) normalized ; compose rigid.
// Writes out[:,384:396] = [rot_new(9), trans_new(3)]
// ---------------------------------------------------------------------------
__global__ void k_quat(const float* __restrict__ act, const float* __restrict__ wq,
                       const float* __restrict__ bq, const float* __restrict__ rot,
                       const float* __restrict__ trans, float* __restrict__ out) {
  const int n = blockIdx.x, t = threadIdx.x;  // 192 = 6 waves, one output each
  const int j = t >> 5, lane = t & 31;
  __shared__ float v6[6];
  float s = 0.f;
  for (int c = lane; c < C1; c += 32) s += act[(size_t)n * C1 + c] * wq[c * 6 + j];
  for (int o = 16; o; o >>= 1) s += __shfl_xor(s, o, 32);
  if (lane == 0) v6[j] = s + bq[j];
  __syncthreads();
  if (t == 0) {
    float a = v6[0], b = v6[1], c = v6[2], w = 1.f;
    const float inv = rsqrtf(w * w + a * a + b * b + c * c);
    w *= inv; a *= inv; b *= inv; c *= inv;
    float U[9];
    U[0] = 1.f - 2.f * (b * b + c * c); U[1] = 2.f * (a * b - w * c); U[2] = 2.f * (a * c + w * b);
    U[3] = 2.f * (a * b + w * c); U[4] = 1.f - 2.f * (a * a + c * c); U[5] = 2.f * (b * c - w * a);
    U[6] = 2.f * (a * c - w * b); U[7] = 2.f * (b * c + w * a); U[8] = 1.f - 2.f * (a * a + b * b);
    const float* R = rot + n * 9;
    const float* T = trans + n * 3;
    float* o = out + (size_t)n * OCOL + C1;
#pragma unroll
    for (int i = 0; i < 3; ++i)
#pragma unroll
      for (int k = 0; k < 3; ++k)
        o[i * 3 + k] = R[i * 3 + 0] * U[k] + R[i * 3 + 1] * U[3 + k] + R[i * 3 + 2] * U[6 + k];
    const float tx = v6[3], ty = v6[4], tz = v6[5];
    o[9]  = R[0] * tx + R[1] * ty + R[2] * tz + T[0];
    o[10] = R[3] * tx + R[4] * ty + R[5] * tz + T[1];
    o[11] = R[6] * tx + R[7] * ty + R[8] * tz + T[2];
  }
}

// ---------------------------------------------------------------------------
static inline int gblocks(int tiles) { return (tiles + 7) / 8; }

extern "C" void kernel_launch(void* const* d_in, const int* in_sizes, int n_in,
                              void* d_out, int out_size, void* d_ws, size_t ws_size,
                              hipStream_t stream) {
  const float* in1d  = (const float*)d_in[0];
  const float* in2d  = (const float*)d_in[1];
  const float* mask  = (const float*)d_in[2];
  const float* rot   = (const float*)d_in[3];
  const float* trans = (const float*)d_in[4];
  const float* w_qp  = (const float*)d_in[5];
  const float* b_qp  = (const float*)d_in[6];
  const float* w_kp  = (const float*)d_in[7];
  const float* b_kp  = (const float*)d_in[8];
  const float* w_vp  = (const float*)d_in[9];
  const float* b_vp  = (const float*)d_in[10];
  const float* w_qs  = (const float*)d_in[11];
  const float* w_ks  = (const float*)d_in[12];
  const float* w_vs  = (const float*)d_in[13];
  const float* tpw   = (const float*)d_in[14];
  const float* w_2d  = (const float*)d_in[15];
  const float* b_2d  = (const float*)d_in[16];
  const float* w_out = (const float*)d_in[17];
  const float* b_out = (const float*)d_in[18];
  const float* ln1s  = (const float*)d_in[19];
  const float* ln1b  = (const float*)d_in[20];
  const float* w_t1  = (const float*)d_in[21];
  const float* b_t1  = (const float*)d_in[22];
  const float* w_t2  = (const float*)d_in[23];
  const float* b_t2  = (const float*)d_in[24];
  const float* w_t3  = (const float*)d_in[25];
  const float* b_t3  = (const float*)d_in[26];
  const float* ln2s  = (const float*)d_in[27];
  const float* ln2b  = (const float*)d_in[28];
  const float* w_qt  = (const float*)d_in[29];
  const float* b_qt  = (const float*)d_in[30];
  float* out = (float*)d_out;

  float* ws = (float*)d_ws;
  size_t o = 0;
  float* qp_raw = ws + o; o += (size_t)NRES * 144;
  float* kp_raw = ws + o; o += (size_t)NRES * 144;
  float* vp_raw = ws + o; o += (size_t)NRES * 288;
  float* qg     = ws + o; o += (size_t)NRES * 144;
  float* kg     = ws + o; o += (size_t)NRES * 144;
  float* vg     = ws + o; o += (size_t)NRES * 288;
  float* qsb    = ws + o; o += (size_t)NRES * 192;
  float* ksb    = ws + o; o += (size_t)NRES * 192;
  float* vsb    = ws + o; o += (size_t)NRES * 192;
  float* qn     = ws + o; o += (size_t)NRES * NH;
  float* kn     = ws + o; o += (size_t)NRES * NH;
  float* logits = ws + o; o += (size_t)NH * NN;
  float* rp_g   = ws + o; o += (size_t)NRES * 288;
  float* finalb = ws + o; o += (size_t)NRES * FCOL;
  float* attn_o = ws + o; o += (size_t)NRES * C1;
  float* act1   = ws + o; o += (size_t)NRES * C1;
  float* t1     = ws + o; o += (size_t)NRES * C1;
  float* t2     = ws + o; o += (size_t)NRES * C1;
  float* t3     = ws + o; o += (size_t)NRES * C1;
  float* act2   = ws + o; o += (size_t)NRES * C1;
  (void)ws_size; (void)in_sizes; (void)n_in; (void)out_size;

  // --- projections (WMMA GEMMs, K=384) ---
  k_gemm<<<gblocks(48 * 9),  256, 0, stream>>>(in1d, w_qp, qp_raw, NRES, 144, C1, b_qp, 0);
  k_gemm<<<gblocks(48 * 9),  256, 0, stream>>>(in1d, w_kp, kp_raw, NRES, 144, C1, b_kp, 0);
  k_gemm<<<gblocks(48 * 18), 256, 0, stream>>>(in1d, w_vp, vp_raw, NRES, 288, C1, b_vp, 0);
  k_gemm<<<gblocks(48 * 12), 256, 0, stream>>>(in1d, w_qs, qsb, NRES, 192, C1, nullptr, 0);
  k_gemm<<<gblocks(48 * 12), 256, 0, stream>>>(in1d, w_ks, ksb, NRES, 192, C1, nullptr, 0);
  k_gemm<<<gblocks(48 * 12), 256, 0, stream>>>(in1d, w_vs, vsb, NRES, 192, C1, nullptr, 0);

  // --- rigid transforms + squared norms ---
  k_rigid_fwd<<<(NRES * NH + 255) / 256, 256, 0, stream>>>(qp_raw, rot, trans, qg, qn, 4);
  k_rigid_fwd<<<(NRES * NH + 255) / 256, 256, 0, stream>>>(kp_raw, rot, trans, kg, kn, 4);
  k_rigid_fwd<<<(NRES * NH + 255) / 256, 256, 0, stream>>>(vp_raw, rot, trans, vg, nullptr, 8);

  // --- logits: 2d bias stream (302 MB, TDM->LDS), qk terms, softmax ---
  k_logits2d<<<NN / 16 / 8, 256, 0, stream>>>(in2d, w_2d, b_2d, logits);
  k_qk_terms<<<dim3(6, 48, NH), 256, 0, stream>>>(qsb, ksb, qg, kg, qn, kn, tpw, mask, logits);
  k_softmax<<<dim3(NRES, NH), 256, 0, stream>>>(logits);

  // --- attention outputs ---
  k_attn_v<<<dim3(18, 1, NH), 256, 0, stream>>>(logits, vsb, vg, finalb, rp_g);
  k_res2d<<<NRES, 256, 0, stream>>>(logits, in2d, finalb);   // second 302 MB stream
  k_rp_local<<<(NRES * NH * PVP + 255) / 256, 256, 0, stream>>>(rp_g, rot, trans, finalb);

  // --- output projection, residual LN, transition MLP, LN ---
  k_gemm<<<gblocks(48 * 24), 256, 0, stream>>>(finalb, w_out, attn_o, NRES, C1, FCOL, b_out, 0);
  k_add_ln<<<NRES, 128, 0, stream>>>(in1d, attn_o, ln1s, ln1b, act1, nullptr);
  k_gemm<<<gblocks(48 * 24), 256, 0, stream>>>(act1, w_t1, t1, NRES, C1, C1, b_t1, 1);
  k_gemm<<<gblocks(48 * 24), 256, 0, stream>>>(t1, w_t2, t2, NRES, C1, C1, b_t2, 1);
  k_gemm<<<gblocks(48 * 24), 256, 0, stream>>>(t2, w_t3, t3, NRES, C1, C1, b_t3, 0);
  k_add_ln<<<NRES, 128, 0, stream>>>(act1, t3, ln2s, ln2b, act2, out);  // also out[:,0:384]

  // --- rigid update ---
  k_quat<<<NRES, 192, 0, stream>>>(act2, w_qt, b_qt, rot, trans, out);
}